// ASTEncoder_62354335204075
// MI455X (gfx1250) — compile-verified
//
#include <hip/hip_runtime.h>
#include <hip/hip_bf16.h>
#include <math.h>

// ---------------------------------------------------------------------------
// GCN encoder for MI455X (gfx1250, wave32).
//  h = emb[x]
//  2x: agg = scatter_add(h over bidirectional edges); a = agg * rsqrt(deg)
//      h = gelu(a @ W + b + h)            <- V_WMMA_F32_16X16X4_F32 path
//  pool mean per graph; out = LN(gfeat @ Wo + bo) * gamma + beta
// ---------------------------------------------------------------------------

#define HDIM 128
#define ODIM 256

typedef __attribute__((ext_vector_type(2))) float v2f;
typedef __attribute__((ext_vector_type(8))) float v8f;

__global__ __launch_bounds__(256) void k_zero(float* __restrict__ p, int n) {
  int i = blockIdx.x * 256 + threadIdx.x;
  if (i < n) p[i] = 0.0f;
}

// h[n][c] = emb[x[n]][c]; total = N*HDIM threads
__global__ __launch_bounds__(256) void k_embed(const int* __restrict__ x,
                                               const float* __restrict__ emb,
                                               float* __restrict__ h, int total) {
  int i = blockIdx.x * 256 + threadIdx.x;
  if (i < total) {
    int node = i >> 7;
    int c = i & (HDIM - 1);
    h[i] = emb[x[node] * HDIM + c];
  }
}

// deg[src]++ and deg[dst]++ per undirected edge (bidirectional message count)
__global__ __launch_bounds__(256) void k_degree(const int* __restrict__ ei,
                                                float* __restrict__ deg, int E) {
  int e = blockIdx.x * 256 + threadIdx.x;
  if (e < E) {
    int s = ei[e];
    int d = ei[E + e];
    unsafeAtomicAdd(&deg[s], 1.0f);
    unsafeAtomicAdd(&deg[d], 1.0f);
  }
}

// in-place: deg -> 1/sqrt(max(deg,1))
__global__ __launch_bounds__(256) void k_scale(float* __restrict__ deg, int N) {
  int i = blockIdx.x * 256 + threadIdx.x;
  if (i < N) deg[i] = 1.0f / sqrtf(fmaxf(deg[i], 1.0f));
}

// one wave lane handles 4 consecutive channels of one edge; both directions
__global__ __launch_bounds__(256) void k_scatter(const int* __restrict__ ei,
                                                 const float* __restrict__ h,
                                                 float* __restrict__ agg, int E) {
  int tid = blockIdx.x * 256 + threadIdx.x;
  int e = tid >> 5;
  if (e >= E) return;
  int c = (tid & 31) * 4;
  int s = ei[e];
  int d = ei[E + e];
  float4 a = *(const float4*)(h + (size_t)s * HDIM + c);
  float4 b = *(const float4*)(h + (size_t)d * HDIM + c);
  float* pd = agg + (size_t)d * HDIM + c;
  float* ps = agg + (size_t)s * HDIM + c;
  unsafeAtomicAdd(pd + 0, a.x);
  unsafeAtomicAdd(pd + 1, a.y);
  unsafeAtomicAdd(pd + 2, a.z);
  unsafeAtomicAdd(pd + 3, a.w);
  unsafeAtomicAdd(ps + 0, b.x);
  unsafeAtomicAdd(ps + 1, b.y);
  unsafeAtomicAdd(ps + 2, b.z);
  unsafeAtomicAdd(ps + 3, b.w);
}

// -------- fused (agg*scale) @ W + b + h -> exact GELU, via fp32 WMMA --------
// block = 16 node rows; wave w of 8 handles output cols [16w, 16w+16)
// K loop: 128/4 = 32 x V_WMMA_F32_16X16X4_F32
__global__ __launch_bounds__(256) void k_gcn_gemm(const float* __restrict__ agg,
                                                  float* __restrict__ h,
                                                  const float* __restrict__ scale,
                                                  const float* __restrict__ W,
                                                  const float* __restrict__ bias,
                                                  int N) {
  const int block_row = blockIdx.x * 16;
  const int wave = threadIdx.x >> 5;
  const int lane = threadIdx.x & 31;
  const int half = lane >> 4;     // 0: K=0,1 | 1: K=2,3 (A frag), M blocks (D frag)
  const int l15 = lane & 15;
  const int col_base = wave * 16;

  // A fragment source: row = block_row + l15, channels kbase + half*2 + {0,1}
  int arow = block_row + l15;
  if (arow >= N) arow = N - 1;                 // N is a multiple of 16; safety clamp
  const float s = scale[arow];
  const float* aptr = agg + (size_t)arow * HDIM + half * 2;
  // B fragment source: W[k + half*2 + {0,1}][col_base + l15]
  const float* bptr = W + (size_t)(half * 2) * HDIM + col_base + l15;

  v8f acc = {0.f, 0.f, 0.f, 0.f, 0.f, 0.f, 0.f, 0.f};
#pragma unroll
  for (int k = 0; k < HDIM; k += 4) {
    v2f a = *(const v2f*)(aptr + k);           // 8B-aligned: row base 512B, +k*4 (+0/8)
    a.x *= s;
    a.y *= s;
    v2f b;
    b.x = bptr[(size_t)k * HDIM];
    b.y = bptr[(size_t)(k + 1) * HDIM];
    acc = __builtin_amdgcn_wmma_f32_16x16x4_f32(
        /*neg_a=*/false, a, /*neg_b=*/false, b,
        /*c_mod=*/(short)0, acc, /*reuse_a=*/false, /*reuse_b=*/false);
  }

  const int col = col_base + l15;
  const float bc = bias[col];
#pragma unroll
  for (int r = 0; r < 8; ++r) {
    int row = block_row + r + half * 8;        // D layout: VGPR r -> M=r (+8 upper half)
    if (row < N) {
      size_t idx = (size_t)row * HDIM + col;
      float xv = acc[r] + bc + h[idx];         // bias + residual
      float g = 0.5f * xv * (1.0f + erff(xv * 0.70710678118654752f));  // exact GELU
      h[idx] = g;                              // in-place: same element read/written
    }
  }
}

// per-(node,channel) atomic accumulate into graph sums; total = N*HDIM
__global__ __launch_bounds__(256) void k_pool(const int* __restrict__ batch,
                                              const float* __restrict__ h,
                                              float* __restrict__ gsum,
                                              float* __restrict__ cnt, int total) {
  int i = blockIdx.x * 256 + threadIdx.x;
  if (i >= total) return;
  int n = i >> 7;
  int c = i & (HDIM - 1);
  int b = batch[n];
  unsafeAtomicAdd(&gsum[b * HDIM + c], h[i]);
  if (c == 0) unsafeAtomicAdd(&cnt[b], 1.0f);
}

// one block per graph: gfeat = gsum/cnt; out = LN(gfeat @ Wo + bo)*gamma+beta
__global__ __launch_bounds__(256) void k_out_ln(const float* __restrict__ gsum,
                                                const float* __restrict__ cnt,
                                                const float* __restrict__ Wo,
                                                const float* __restrict__ bo,
                                                const float* __restrict__ gamma,
                                                const float* __restrict__ beta,
                                                float* __restrict__ out) {
  __shared__ float gfeat[HDIM];
  __shared__ float red[ODIM];
  const int g = blockIdx.x;
  const int t = threadIdx.x;

  if (t < HDIM) {
    float cv = fmaxf(cnt[g], 1.0f);
    gfeat[t] = gsum[g * HDIM + t] / cv;
  }
  __syncthreads();

  float acc = bo[t];
#pragma unroll
  for (int k = 0; k < HDIM; ++k) acc += gfeat[k] * Wo[(size_t)k * ODIM + t];

  // mean over 256
  red[t] = acc;
  __syncthreads();
  for (int sw = ODIM / 2; sw > 0; sw >>= 1) {
    if (t < sw) red[t] += red[t + sw];
    __syncthreads();
  }
  float mu = red[0] * (1.0f / ODIM);
  __syncthreads();

  float dv = acc - mu;
  red[t] = dv * dv;
  __syncthreads();
  for (int sw = ODIM / 2; sw > 0; sw >>= 1) {
    if (t < sw) red[t] += red[t + sw];
    __syncthreads();
  }
  float var = red[0] * (1.0f / ODIM);            // population var, matches jnp.var
  __syncthreads();

  out[(size_t)g * ODIM + t] = dv * rsqrtf(var + 1e-5f) * gamma[t] + beta[t];
}

// ---------------------------------------------------------------------------
extern "C" void kernel_launch(void* const* d_in, const int* in_sizes, int n_in,
                              void* d_out, int out_size, void* d_ws, size_t ws_size,
                              hipStream_t stream) {
  const int* x     = (const int*)d_in[0];
  const int* ei    = (const int*)d_in[1];     // [2, E] row-major: src then dst
  const int* batch = (const int*)d_in[2];
  const float* emb = (const float*)d_in[4];
  const float* W0  = (const float*)d_in[5];
  const float* b0  = (const float*)d_in[6];
  const float* W1  = (const float*)d_in[7];
  const float* b1  = (const float*)d_in[8];
  const float* Wo  = (const float*)d_in[9];
  const float* bo  = (const float*)d_in[10];
  const float* gam = (const float*)d_in[11];
  const float* bet = (const float*)d_in[12];
  float* out = (float*)d_out;

  const int N = in_sizes[0];
  const int E = in_sizes[1] / 2;
  const int B = out_size / ODIM;
  const int NH = N * HDIM;

  // workspace carve-out (256B aligned)
  char* ws = (char*)d_ws;
  size_t off = 0;
  auto carve = [&](size_t bytes) {
    char* p = ws + off;
    off = (off + bytes + 255) & ~(size_t)255;
    return p;
  };
  float* scale = (float*)carve((size_t)N * 4);       // deg, then rsqrt(deg)
  float* agg   = (float*)carve((size_t)NH * 4);
  float* h     = (float*)carve((size_t)NH * 4);
  float* gsum  = (float*)carve((size_t)B * HDIM * 4);
  float* cnt   = (float*)carve((size_t)B * 4);
  (void)ws_size;
  (void)n_in;

  const int TPB = 256;
  auto blks = [](int n, int t) { return (n + t - 1) / t; };

  // zero accumulators (re-done every launch: graph-replay safe)
  k_zero<<<blks(N, TPB), TPB, 0, stream>>>(scale, N);
  k_zero<<<blks(B * HDIM, TPB), TPB, 0, stream>>>(gsum, B * HDIM);
  k_zero<<<blks(B, TPB), TPB, 0, stream>>>(cnt, B);

  k_embed<<<blks(NH, TPB), TPB, 0, stream>>>(x, emb, h, NH);
  k_degree<<<blks(E, TPB), TPB, 0, stream>>>(ei, scale, E);
  k_scale<<<blks(N, TPB), TPB, 0, stream>>>(scale, N);

  const int gemm_blocks = (N + 15) / 16;
  const int scat_threads = E * 32;

  // layer 0
  k_zero<<<blks(NH, TPB), TPB, 0, stream>>>(agg, NH);
  k_scatter<<<blks(scat_threads, TPB), TPB, 0, stream>>>(ei, h, agg, E);
  k_gcn_gemm<<<gemm_blocks, TPB, 0, stream>>>(agg, h, scale, W0, b0, N);

  // layer 1
  k_zero<<<blks(NH, TPB), TPB, 0, stream>>>(agg, NH);
  k_scatter<<<blks(scat_threads, TPB), TPB, 0, stream>>>(ei, h, agg, E);
  k_gcn_gemm<<<gemm_blocks, TPB, 0, stream>>>(agg, h, scale, W1, b1, N);

  // pooling + output projection + layernorm
  k_pool<<<blks(NH, TPB), TPB, 0, stream>>>(batch, h, gsum, cnt, NH);
  k_out_ln<<<B, ODIM, 0, stream>>>(gsum, cnt, Wo, bo, gam, bet, out);
}